// GCN_75273596830206
// MI455X (gfx1250) — compile-verified
//
#include <hip/hip_runtime.h>
#include <math.h>

typedef __attribute__((ext_vector_type(2))) float v2f;
typedef __attribute__((ext_vector_type(8))) float v8f;

#define D_IN  64
#define D_HID 64
#define D_OUT 16

// ---------------------------------------------------------------- utilities

__global__ void zero_f32(float* __restrict__ p, long long n) {
    long long i = (long long)blockIdx.x * blockDim.x + threadIdx.x;
    if (i < n) p[i] = 0.0f;
}

__global__ void deg_count(const int* __restrict__ dst, float* __restrict__ deg, int E) {
    int e = blockIdx.x * blockDim.x + threadIdx.x;
    if (e < E) atomicAdd(&deg[dst[e]], 1.0f);
}

__global__ void deg_to_dinv(float* __restrict__ deg, int n) {
    int i = blockIdx.x * blockDim.x + threadIdx.x;
    if (i < n) deg[i] = rsqrtf(deg[i] + 1.0f);   // deg includes self-loop (+1)
}

// ------------------------------------------------- WMMA GEMM 1: h1 = x @ W1
// grid.x = N/16 (N % 16 == 0), block = 128 threads = 4 waves.
// Wave w computes the 16x16 tile (rows row0..row0+15, cols 16w..16w+15).
// f32 WMMA 16x16x4 layouts (ISA 7.12.2):
//   A 16x4 : lane L -> M = L&15, holds K = 2*(L>>4)+{0,1}
//   B 4x16 : lane L -> N = L&15, holds K = 2*(L>>4)+{0,1}
//   C 16x16: VGPR r, lane L -> M = r + 8*(L>>4), N = L&15
__global__ void gemm_x_w1(const float* __restrict__ x, const float* __restrict__ W,
                          float* __restrict__ h, int N) {
    const int lane = threadIdx.x & 31;
    const int wave = threadIdx.x >> 5;
    const int l16  = lane & 15;
    const int hb   = lane >> 4;
    const int kb   = hb * 2;
    const int row0 = blockIdx.x * 16;
    const int col0 = wave * 16;
    if (row0 + 16 > N) return;  // wave-uniform guard (never taken for N%16==0)

    v8f acc = {};
    const float* ap = x + (size_t)(row0 + l16) * D_IN + kb;          // A[m][kb..kb+1]
    const float* bp = W + (size_t)kb * D_HID + col0 + l16;           // B[kb..][n]
#pragma unroll
    for (int k0 = 0; k0 < D_IN; k0 += 4) {
        v2f a, b;
        a.x = ap[0];
        a.y = ap[1];
        b.x = bp[0];
        b.y = bp[D_HID];
        acc = __builtin_amdgcn_wmma_f32_16x16x4_f32(false, a, false, b,
                                                    (short)0, acc, false, false);
        ap += 4;
        bp += 4 * D_HID;
    }
#pragma unroll
    for (int r = 0; r < 8; ++r)
        h[(size_t)(row0 + r + 8 * hb) * D_HID + col0 + l16] = acc[r];
}

// ------------------------------------------------ WMMA GEMM 2: h2 = hrelu @ W2
// grid.x = NPAD/128, block = 256 threads = 8 waves; wave w -> rows blk*128+16w.
// All NPAD rows of hrelu are initialized (pad rows are zero), so no predication.
__global__ void gemm_h_w2(const float* __restrict__ h, const float* __restrict__ W,
                          float* __restrict__ o) {
    const int lane = threadIdx.x & 31;
    const int wave = threadIdx.x >> 5;
    const int l16  = lane & 15;
    const int hb   = lane >> 4;
    const int kb   = hb * 2;
    const int row0 = blockIdx.x * 128 + wave * 16;

    v8f acc = {};
    const float* ap = h + (size_t)(row0 + l16) * D_HID + kb;
    const float* bp = W + (size_t)kb * D_OUT + l16;
#pragma unroll
    for (int k0 = 0; k0 < D_HID; k0 += 4) {
        v2f a, b;
        a.x = ap[0];
        a.y = ap[1];
        b.x = bp[0];
        b.y = bp[D_OUT];
        acc = __builtin_amdgcn_wmma_f32_16x16x4_f32(false, a, false, b,
                                                    (short)0, acc, false, false);
        ap += 4;
        bp += 4 * D_OUT;
    }
#pragma unroll
    for (int r = 0; r < 8; ++r)
        o[(size_t)(row0 + r + 8 * hb) * D_OUT + l16] = acc[r];
}

// ----------------------------------------------------------- edge scattering
template <int F>
__global__ void scatter_agg(const int* __restrict__ src, const int* __restrict__ dst,
                            const float* __restrict__ dinv, const float* __restrict__ h,
                            float* __restrict__ agg, long long total) {
    long long t = (long long)blockIdx.x * blockDim.x + threadIdx.x;
    if (t >= total) return;
    const int f = (int)(t & (F - 1));
    const long long e = t >> (F == 64 ? 6 : 4);
    const int s = src[e];
    const int d = dst[e];
    const float w = dinv[s] * dinv[d];
    atomicAdd(&agg[(size_t)d * F + f], h[(size_t)s * F + f] * w);
}

// --------------------------------------------- self-loop + bias + ReLU (L1)
__global__ void self_bias_relu(const float* __restrict__ h1, const float* __restrict__ dinv,
                               const float* __restrict__ b1, float* __restrict__ agg,
                               long long total) {
    long long t = (long long)blockIdx.x * blockDim.x + threadIdx.x;
    if (t >= total) return;
    const int f = (int)(t & (D_HID - 1));
    const long long i = t >> 6;
    const float di = dinv[i];
    const float v = agg[t] + h1[t] * di * di + b1[f];
    agg[t] = v > 0.0f ? v : 0.0f;
}

// --------------------------------- self-loop + bias + log_softmax (layer 2)
__global__ void final_logsoftmax(const float* __restrict__ h2, const float* __restrict__ agg2,
                                 const float* __restrict__ dinv, const float* __restrict__ b2,
                                 float* __restrict__ out, int N) {
    int i = blockIdx.x * blockDim.x + threadIdx.x;
    if (i >= N) return;
    const float di  = dinv[i];
    const float di2 = di * di;
    float v[D_OUT];
    float m = -3.4e38f;
#pragma unroll
    for (int f = 0; f < D_OUT; ++f) {
        v[f] = agg2[(size_t)i * D_OUT + f] + h2[(size_t)i * D_OUT + f] * di2 + b2[f];
        m = fmaxf(m, v[f]);
    }
    float s = 0.0f;
#pragma unroll
    for (int f = 0; f < D_OUT; ++f) s += expf(v[f] - m);
    const float lse = m + logf(s);
#pragma unroll
    for (int f = 0; f < D_OUT; ++f) out[(size_t)i * D_OUT + f] = v[f] - lse;
}

// ------------------------------------------------------------------- launch
extern "C" void kernel_launch(void* const* d_in, const int* in_sizes, int n_in,
                              void* d_out, int out_size, void* d_ws, size_t ws_size,
                              hipStream_t stream) {
    const float* x  = (const float*)d_in[0];
    const int*   ei = (const int*)d_in[1];
    const float* W1 = (const float*)d_in[2];
    const float* b1 = (const float*)d_in[3];
    const float* W2 = (const float*)d_in[4];
    const float* b2 = (const float*)d_in[5];
    float* out = (float*)d_out;

    const int N = in_sizes[0] / D_IN;       // 100000
    const int E = in_sizes[1] / 2;          // 1600000
    const int NPAD = ((N + 127) / 128) * 128;

    const int* src = ei;
    const int* dst = ei + E;

    float* ws   = (float*)d_ws;
    float* deg  = ws;                                  // [NPAD]       -> becomes dinv
    float* agg1 = deg  + NPAD;                         // [NPAD*64]
    float* agg2 = agg1 + (size_t)NPAD * D_HID;         // [NPAD*16]
    float* h1   = agg2 + (size_t)NPAD * D_OUT;         // [NPAD*64]
    float* h2   = h1   + (size_t)NPAD * D_HID;         // [NPAD*16] (fully overwritten)

    // Zero deg | agg1 | agg2 | h1 (contiguous prefix of ws).
    const long long zc = (long long)NPAD * (1 + D_HID + D_OUT + D_HID);
    zero_f32<<<(int)((zc + 255) / 256), 256, 0, stream>>>(ws, zc);

    // Degrees -> dinv (in place).
    deg_count<<<(E + 255) / 256, 256, 0, stream>>>(dst, deg, E);
    deg_to_dinv<<<(N + 255) / 256, 256, 0, stream>>>(deg, N);

    // Layer 1: h1 = x @ W1 (WMMA f32), scatter, self-loop + bias + ReLU.
    gemm_x_w1<<<N / 16, 128, 0, stream>>>(x, W1, h1, N);
    const long long t1 = (long long)E * D_HID;
    scatter_agg<D_HID><<<(int)((t1 + 255) / 256), 256, 0, stream>>>(src, dst, deg, h1, agg1, t1);
    const long long t2 = (long long)N * D_HID;
    self_bias_relu<<<(int)((t2 + 255) / 256), 256, 0, stream>>>(h1, deg, b1, agg1, t2);

    // Layer 2: h2 = relu(agg1) @ W2 (WMMA f32), scatter, log-softmax.
    gemm_h_w2<<<NPAD / 128, 256, 0, stream>>>(agg1, W2, h2);
    const long long t3 = (long long)E * D_OUT;
    scatter_agg<D_OUT><<<(int)((t3 + 255) / 256), 256, 0, stream>>>(src, dst, deg, h2, agg2, t3);
    final_logsoftmax<<<(N + 255) / 256, 256, 0, stream>>>(h2, agg2, deg, b2, out, N);
}